// ColumnUniform_25744033972461
// MI455X (gfx1250) — compile-verified
//
#include <hip/hip_runtime.h>
#include <cstdint>

// ---------------------------------------------------------------------------
// ColumnUniform: out[e] = (1 / segment_sum(attr, row)[col[e]]) * attr[e]
//
// Memory-bound (≈400 MB HBM ≈ 17 us at 23.3 TB/s; ~0.04 GFLOP). WMMA is
// inapplicable; the CDNA5 features that matter are the async global->LDS DMA
// path (ASYNCcnt) for streaming the edge arrays, native global_atomic_add_f32
// into the L2-resident 4 MB rowsum table (192 MB L2), and non-temporal hints
// on the 240 MB of pure streaming traffic.
// ---------------------------------------------------------------------------

#define TILE 1024  // edges per block in the scatter pass (256 thr * 4)

// Guaranteed-hardware f32 atomic add (no CAS fallback): no-return GVS form,
// uniform SGPR base + per-lane VGPR byte offset. Tracked by STOREcnt;
// s_endpgm's implicit wait-idle covers completion.
__device__ __forceinline__ void atomic_add_f32_hw(float* base, int idx, float v) {
  unsigned byteOff = (unsigned)idx * 4u;
  asm volatile("global_atomic_add_f32 %0, %1, %2"
               :: "v"(byteOff), "v"(v), "s"(base)
               : "memory");
}

__global__ void cu_zero(float* __restrict__ buf, const int* __restrict__ Np) {
  const int n = *Np;
  for (int i = blockIdx.x * blockDim.x + threadIdx.x; i < n;
       i += gridDim.x * blockDim.x)
    buf[i] = 0.0f;
}

__global__ __launch_bounds__(256) void cu_scatter(
    const int* __restrict__ row, const float* __restrict__ attr,
    float* __restrict__ rowsum, int E) {
  __shared__ float sAttr[TILE];
  __shared__ int   sRow[TILE];
  const int tid    = threadIdx.x;
  const int mybase = blockIdx.x * TILE + tid * 4;

  if (mybase + 4 <= E) {
    // Generic pointers to __shared__ carry the LDS byte offset in their low
    // 32 bits (SHARED aperture), which is exactly what the async-to-LDS VDST
    // operand wants.
    unsigned ldsA = (unsigned)(uintptr_t)&sAttr[tid * 4];
    unsigned ldsR = (unsigned)(uintptr_t)&sRow[tid * 4];
    const float* ga = attr + mybase;
    const int*   gr = row + mybase;
    // Each lane DMAs 16B of attr and 16B of row into its own LDS slots.
    asm volatile("global_load_async_to_lds_b128 %0, %1, off"
                 :: "v"(ldsA), "v"(ga) : "memory");
    asm volatile("global_load_async_to_lds_b128 %0, %1, off"
                 :: "v"(ldsR), "v"(gr) : "memory");
    asm volatile("s_wait_asynccnt 0" ::: "memory");
#pragma unroll
    for (int k = 0; k < 4; ++k)
      atomic_add_f32_hw(rowsum, sRow[tid * 4 + k], sAttr[tid * 4 + k]);
  } else if (mybase < E) {
    const int end = (mybase + 4 < E) ? (mybase + 4) : E;
    for (int e = mybase; e < end; ++e)
      atomic_add_f32_hw(rowsum, row[e], attr[e]);
  }
}

__global__ void cu_recip(float* __restrict__ rs, const int* __restrict__ Np) {
  const int n = *Np;
  for (int i = blockIdx.x * blockDim.x + threadIdx.x; i < n;
       i += gridDim.x * blockDim.x)
    rs[i] = 1.0f / rs[i];  // exact IEEE divide, matches jnp 1.0/rowsum
}

__global__ void cu_gather(const int* __restrict__ col,
                          const float* __restrict__ attr,
                          const float* __restrict__ inv,
                          float* __restrict__ out, int E) {
  const int e = blockIdx.x * blockDim.x + threadIdx.x;
  if (e < E) {
    // NT on the 240 MB of streaming traffic; inv[] stays hot in L2.
    const int   c = __builtin_nontemporal_load(col + e);
    const float a = __builtin_nontemporal_load(attr + e);
    __builtin_nontemporal_store(inv[c] * a, out + e);
  }
}

extern "C" void kernel_launch(void* const* d_in, const int* in_sizes, int n_in,
                              void* d_out, int out_size, void* d_ws,
                              size_t ws_size, hipStream_t stream) {
  (void)n_in; (void)out_size; (void)ws_size;
  const int*   edge_index = (const int*)d_in[0];   // [2, E] int32
  const float* attr       = (const float*)d_in[1]; // [E] f32
  const int*   Np         = (const int*)d_in[2];   // scalar N on device
  const int    E          = in_sizes[1];
  const int*   row        = edge_index;            // edge_index[0]
  const int*   col        = edge_index + E;        // edge_index[1]
  float*       rowsum     = (float*)d_ws;          // N floats (4 MB)
  float*       out        = (float*)d_out;

  cu_zero   <<<2048, 256, 0, stream>>>(rowsum, Np);
  cu_scatter<<<(E + TILE - 1) / TILE, 256, 0, stream>>>(row, attr, rowsum, E);
  cu_recip  <<<2048, 256, 0, stream>>>(rowsum, Np);
  cu_gather <<<(E + 255) / 256, 256, 0, stream>>>(col, attr, rowsum, out, E);
}